// TiledWaveletVolume_61993557950728
// MI455X (gfx1250) — compile-verified
//
#include <hip/hip_runtime.h>
#include <hip/hip_bf16.h>

typedef __attribute__((ext_vector_type(2))) float v2f;
typedef __attribute__((ext_vector_type(4))) float v4f;
typedef __attribute__((ext_vector_type(8))) float v8f;

#define C3 0.35355339059327373f  // (1/sqrt(2))^3

// ---------------------------------------------------------------------------
// Constant per-lane A-fragment table for D = blockdiag(M8,M8) x B via four
// chained V_WMMA_F32_16X16X4_F32. M8[p][b] = C3 * (-1)^popc(b&p).
// Lane layout (ISA 7.12.2, 32-bit A 16x4): lanes 0-15 hold M=0..15 with
// K = 4m,4m+1 ; lanes 16-31 hold M=0..15 with K = 4m+2,4m+3.
// ---------------------------------------------------------------------------
constexpr int cpopc(int x) {
    int c = 0;
    while (x) { c += x & 1; x >>= 1; }
    return c;
}

struct alignas(16) ATab {
    float v[32][8];  // [lane][2*m + i]
};

constexpr ATab makeATab() {
    ATab t{};
    for (int lane = 0; lane < 32; ++lane) {
        const int row = lane & 15;   // M row
        const int half = lane >> 4;  // which K sub-rows this half-wave holds
        for (int m = 0; m < 4; ++m) {
            for (int i = 0; i < 2; ++i) {
                const int k = 4 * m + 2 * half + i;
                const int p = row & 7, tr = row >> 3;
                const int b = k & 7, tk = k >> 3;
                t.v[lane][2 * m + i] =
                    (tr != tk) ? 0.0f : ((cpopc(b & p) & 1) ? -C3 : C3);
            }
        }
    }
    return t;
}

__constant__ ATab g_atab = makeATab();

// ---------------------------------------------------------------------------
// One inverse-Haar level.
//   approx : [8][R][R][R]      (channel-major)
//   detail : [7][8][R][R][R]   (channel-major per subband)
//   out    : INTERLEAVED ? [2R][2R][2R][8] : [8][2R][2R][2R]
// Block = 256 threads = 8 waves. Each wave: 16 coarse-X columns x 1 channel
// pair. grid = (R/32, R, R). R is a template constant so all strides fold
// into shifts.
// ---------------------------------------------------------------------------
template <int R, bool INTERLEAVED>
__global__ __launch_bounds__(256) void haar_synth_kernel(
    const float* __restrict__ approx, const float* __restrict__ detail,
    float* __restrict__ out) {
    const int lane = threadIdx.x & 31;
    const int wave = threadIdx.x >> 5;
    const int g    = wave >> 2;        // which 16-wide X tile (0,1)
    const int c0   = (wave & 3) * 2;   // channel pair base
    const int Z = blockIdx.z, Y = blockIdx.y;
    const int X0 = blockIdx.x * 32 + g * 16;
    const int N  = lane & 15;          // WMMA column -> coarse X offset
    const int X  = X0 + N;
    const int half = lane >> 4;
    constexpr int R2 = 2 * R;

    // A fragments from the constant table (4x 8-byte loads per lane).
    const v2f* at = (const v2f*)g_atab.v[lane];
    v2f afrag[4];
#pragma unroll
    for (int m = 0; m < 4; ++m) afrag[m] = at[m];

    // B fragments: rows k = subband b + 8*t (t selects channel of the pair),
    // column N -> coarse X. Subband 0 = approx, 1..7 = detail[b-1].
    v2f bfrag[4];
#pragma unroll
    for (int m = 0; m < 4; ++m) {
#pragma unroll
        for (int i = 0; i < 2; ++i) {
            const int k = 4 * m + 2 * half + i;
            const int b = k & 7;
            const int c = c0 + (k >> 3);
            float v;
            if (b == 0) {
                v = approx[(((size_t)c * R + Z) * R + Y) * R + X];
            } else {
                v = detail[((((size_t)(b - 1) * 8 + c) * R + Z) * R + Y) * R + X];
            }
            if (i == 0) bfrag[m].x = v; else bfrag[m].y = v;
        }
    }

    v8f d = {};
#pragma unroll
    for (int m = 0; m < 4; ++m) {
        // (neg_a, A, neg_b, B, c_mod, C, reuse_a, reuse_b)
        d = __builtin_amdgcn_wmma_f32_16x16x4_f32(
            false, afrag[m], false, bfrag[m], (short)0, d, false, false);
    }

    // D VGPR r = output position p=r (bit0=px, bit1=py, bit2=pz);
    // lanes 0-15 -> channel c0, lanes 16-31 -> channel c0+1; column N -> X.
    if constexpr (!INTERLEAVED) {
        const int c = c0 + half;
#pragma unroll
        for (int r = 0; r < 8; ++r) {
            const int x = 2 * X + (r & 1);
            const int y = 2 * Y + ((r >> 1) & 1);
            const int z = 2 * Z + (r >> 2);
            out[(((size_t)c * R2 + z) * R2 + y) * R2 + x] = d[r];
        }
    } else {
        // Stage through LDS, then push fully-coalesced interleaved 32B chunks
        // straight from LDS to memory with the CDNA5 async store path.
        __shared__ float tile[2][2][2][32][8];  // [g][pz][py][x][c] = 4 KB
        const int c = c0 + half;
#pragma unroll
        for (int r = 0; r < 8; ++r) {
            tile[g][r >> 2][(r >> 1) & 1][2 * N + (r & 1)][c] = d[r];
        }
        __syncthreads();
        const int tid = threadIdx.x;
        const int g2 = tid >> 7;
        const int i  = tid & 127;
        const int pz = i >> 6, py = (i >> 5) & 1, x = i & 31;
        const int z = 2 * Z + pz, y = 2 * Y + py;
        const int gx = 2 * (blockIdx.x * 32 + g2 * 16) + x;
        float* dst = &out[((((size_t)z * R2) + y) * R2 + gx) * 8];
        // Generic pointer to a __shared__ object: low 32 bits are the LDS
        // byte offset (LDS aperture addressing).
        const unsigned ldsoff =
            (unsigned)(uintptr_t)&tile[g2][pz][py][x][0];
        const unsigned long long gaddr = (unsigned long long)(uintptr_t)dst;
        // ASYNCcnt-tracked LDS -> global stores; offset: advances both the
        // global and the LDS address per the ISA async addressing rules.
        asm volatile(
            "global_store_async_from_lds_b128 %0, %1, off\n\t"
            "global_store_async_from_lds_b128 %0, %1, off offset:16"
            :: "v"(gaddr), "v"(ldsoff) : "memory");
        asm volatile("s_wait_asynccnt 0x0" ::: "memory");
    }
}

// ---------------------------------------------------------------------------
// Trilinear sampling of the interleaved [256][256][256][8] volume.
// One thread per point: 8 corners x 32B aligned gathers, FMA blend.
// ---------------------------------------------------------------------------
__global__ __launch_bounds__(256) void trilerp_kernel(
    const float* __restrict__ xyz, const float* __restrict__ vol,
    float* __restrict__ out, int n) {
    const int i = blockIdx.x * blockDim.x + threadIdx.x;
    if (i >= n) return;
    const float px = xyz[3 * (size_t)i + 0];
    const float py = xyz[3 * (size_t)i + 1];
    const float pz = xyz[3 * (size_t)i + 2];
    // g = clip((p/1.5 + 1) * 0.5 * 255, 0, 255) = clip(p*85 + 127.5, 0, 255)
    const float gx = fminf(fmaxf(px * 85.0f + 127.5f, 0.0f), 255.0f);
    const float gy = fminf(fmaxf(py * 85.0f + 127.5f, 0.0f), 255.0f);
    const float gz = fminf(fmaxf(pz * 85.0f + 127.5f, 0.0f), 255.0f);
    const float x0f = floorf(gx), y0f = floorf(gy), z0f = floorf(gz);
    const float fx = gx - x0f, fy = gy - y0f, fz = gz - z0f;
    const int x0 = (int)x0f, y0 = (int)y0f, z0 = (int)z0f;
    const int x1 = min(x0 + 1, 255), y1 = min(y0 + 1, 255), z1 = min(z0 + 1, 255);
    const float wx[2] = {1.0f - fx, fx};
    const float wy[2] = {1.0f - fy, fy};
    const float wz[2] = {1.0f - fz, fz};
    const int xi[2] = {x0, x1}, yi[2] = {y0, y1}, zi[2] = {z0, z1};

    const v4f* v4 = (const v4f*)vol;
    v4f accLo = {}, accHi = {};
#pragma unroll
    for (int dz = 0; dz < 2; ++dz) {
#pragma unroll
        for (int dy = 0; dy < 2; ++dy) {
#pragma unroll
            for (int dx = 0; dx < 2; ++dx) {
                const float w = wz[dz] * wy[dy] * wx[dx];
                const size_t base =
                    ((((size_t)zi[dz] * 256) + yi[dy]) * 256 + xi[dx]) * 2;
                accLo += w * v4[base];
                accHi += w * v4[base + 1];
            }
        }
    }
    v4f* o = (v4f*)(out + (size_t)i * 8);
    o[0] = accLo;
    o[1] = accHi;
}

extern "C" void kernel_launch(void* const* d_in, const int* in_sizes, int n_in,
                              void* d_out, int out_size, void* d_ws, size_t ws_size,
                              hipStream_t stream) {
    const float* xyz    = (const float*)d_in[0];  // (1e6, 3)
    const float* approx = (const float*)d_in[1];  // (1, 8, 32,32,32)
    const float* det0   = (const float*)d_in[2];  // (7, 8, 32,32,32)
    const float* det1   = (const float*)d_in[3];  // (7, 8, 64,64,64)
    const float* det2   = (const float*)d_in[4];  // (7, 8, 128,128,128)
    float* out = (float*)d_out;

    float* ws = (float*)d_ws;
    float* vol64  = ws;                                   // [8][64^3]  ch-major
    float* vol128 = vol64 + (size_t)8 * 64 * 64 * 64;     // [8][128^3] ch-major
    float* vol256 = vol128 + (size_t)8 * 128 * 128 * 128; // [256^3][8] interleaved

    const dim3 blk(256);
    haar_synth_kernel<32, false><<<dim3(1, 32, 32), blk, 0, stream>>>(approx, det0, vol64);
    haar_synth_kernel<64, false><<<dim3(2, 64, 64), blk, 0, stream>>>(vol64, det1, vol128);
    haar_synth_kernel<128, true><<<dim3(4, 128, 128), blk, 0, stream>>>(vol128, det2, vol256);

    const int n = in_sizes[0] / 3;
    trilerp_kernel<<<(n + 255) / 256, 256, 0, stream>>>(xyz, vol256, out, n);
}